// GravityNet_41463614276194
// MI455X (gfx1250) — compile-verified
//
#include <hip/hip_runtime.h>
#include <stdint.h>

// N-body gravity, N=2048, 2D, 256 steps, fp32 (matches reference).
// - Per-step pairwise kernel: packed (x,y,m) staged to LDS via CDNA5 async
//   global->LDS b128 loads (ASYNCcnt); 64 blocks x 8 wave32; lane=i-body,
//   wave=j-chunk; one ds_load_b128 per pair interaction.
// - Indicator reduction (ke/pe over 64 block-partials per step) done with
//   chained V_WMMA_F32_16X16X4_F32 (matrix x ones-vector).

#define NB     2048
#define STEPS  256
#define ROWF   (NB * 4 + 2)     // 8194 floats per output row
#define DT     0.02f
#define NBLK   64               // blocks in the pairwise kernel
#define TPB    256              // threads per block (8 wave32)
#define TILE_I 32               // i-bodies per block (== lanes)
#define JCH    8                // j-chunks per block (== waves)
#define JLEN   (NB / JCH)       // 256 j per chunk

// Low 32 bits of a flat pointer into __shared__ == LDS byte address
// (ISA 10.2: generic LDS address = {aperture, lds_offset[31:0]}).
__device__ __forceinline__ uint32_t lds_addr32(const void* p) {
  return (uint32_t)(uintptr_t)p;
}

// CDNA5 async copy global -> LDS (ASYNCcnt), CDNA5 ISA.
__device__ __forceinline__ void async_copy_b128(uint32_t lds, const void* g) {
  asm volatile("global_load_async_to_lds_b128 %0, %1, off"
               :: "v"(lds), "v"(g) : "memory");
}
__device__ __forceinline__ void async_wait_all() {
  asm volatile("s_wait_asynccnt 0x0" ::: "memory");
}

__global__ __launch_bounds__(32) void grav_init(
    const float* __restrict__ x0, const float* __restrict__ v0,
    const float* __restrict__ m, float* __restrict__ row0,
    float4* __restrict__ pxm0, float2* __restrict__ vb0,
    float* __restrict__ partKE0)
{
  const int lane = threadIdx.x;            // one wave per block
  const int i = blockIdx.x * 32 + lane;
  const float xi = x0[2 * i], yi = x0[2 * i + 1];
  const float vx = v0[2 * i], vy = v0[2 * i + 1];
  const float mi = m[i];
  float4 p4; p4.x = xi; p4.y = yi; p4.z = mi; p4.w = 0.f;
  pxm0[i] = p4;                            // packed (x,y,m) image, 16B aligned
  float2 vv; vv.x = vx; vv.y = vy;
  vb0[i] = vv;
  float2* row2 = (float2*)row0;            // rows are 8B-aligned -> float2
  float2 p; p.x = xi; p.y = yi;
  row2[2 * i] = p;
  row2[2 * i + 1] = vv;
  float ke = 0.5f * mi * (vx * vx + vy * vy);
  #pragma unroll
  for (int off = 16; off > 0; off >>= 1) ke += __shfl_down(ke, off, 32);
  if (lane == 0) partKE0[blockIdx.x] = ke;
}

template <bool CHECK_DIAG>
__device__ __forceinline__ void pair_loop(
    const float4* __restrict__ sxm, int j0, int iglob,
    float xi, float yi, float& ax, float& ay, float& pe)
{
  #pragma unroll 8
  for (int jj = 0; jj < JLEN; ++jj) {
    const int j = j0 + jj;
    const float4 bj = sxm[j];              // broadcast ds_load_b128 (x,y,m)
    const float dx = bj.x - xi;
    const float dy = bj.y - yi;
    const float d2 = fmaf(dy, dy, dx * dx);
    float rinv, w1;
    if (CHECK_DIAG) {
      const bool diag = (j == iglob);
      rinv = __builtin_amdgcn_rsqf(diag ? 1.0f : d2);   // v_rsq_f32
      w1 = diag ? 0.0f : bj.z * rinv;                   // m_j / d
    } else {
      rinv = __builtin_amdgcn_rsqf(d2);
      w1 = bj.z * rinv;
    }
    pe += w1;
    const float w3 = w1 * (rinv * rinv);                // m_j / d^3
    ax = fmaf(w3, dx, ax);
    ay = fmaf(w3, dy, ay);
  }
}

__global__ __launch_bounds__(TPB) void grav_step(
    const float4* __restrict__ pxm_prev,   // [NB] packed (x,y,m) at s-1
    float4* __restrict__ pxm_next,         // [NB] packed (x,y,m) at s
    const float2* __restrict__ v_prev,     // [NB] velocities at s-1
    float2* __restrict__ v_next,           // [NB] velocities at s
    float* __restrict__ row_out,           // d_out row s
    const float* __restrict__ Gp,
    float* __restrict__ partKE_s,          // [NBLK] KE partials for step s
    float* __restrict__ partPE_sm1,        // [NBLK] PE partials for step s-1
    int do_update)
{
  __shared__ float4 sxm[NB];               // 32 KB: (x,y,m) per body
  __shared__ float  r_ax[JCH][TILE_I];
  __shared__ float  r_ay[JCH][TILE_I];
  __shared__ float  r_pe[JCH][TILE_I];

  const int tid = threadIdx.x;

  // Stage full packed image into LDS via CDNA5 async path (8 x b128/thread).
  #pragma unroll
  for (int k = 0; k < NB / TPB; ++k) {
    const int idx = tid + k * TPB;
    async_copy_b128(lds_addr32(&sxm[idx]), (const void*)(pxm_prev + idx));
  }
  async_wait_all();                        // drain this wave's ASYNCcnt
  __syncthreads();                         // LDS visible block-wide

  const int lane = tid & 31;               // body within tile
  const int wave = tid >> 5;               // j-chunk
  const int i = blockIdx.x * TILE_I + lane;
  const float4 bi = sxm[i];
  const float xi = bi.x, yi = bi.y;
  float ax = 0.f, ay = 0.f, pe = 0.f;
  const int j0 = wave * JLEN;
  // Block b's i-tile [32b,32b+32) lies entirely inside j-chunk (b>>3):
  // only that wave needs the diagonal guard.
  if (wave == (int)(blockIdx.x >> 3)) {
    pair_loop<true>(sxm, j0, i, xi, yi, ax, ay, pe);
  } else {
    pair_loop<false>(sxm, j0, i, xi, yi, ax, ay, pe);
  }
  r_ax[wave][lane] = ax;
  r_ay[wave][lane] = ay;
  r_pe[wave][lane] = pe;
  __syncthreads();

  if (wave == 0) {                         // wave 0 finishes the 32 bodies
    float sax = 0.f, say = 0.f, spe = 0.f;
    #pragma unroll
    for (int c = 0; c < JCH; ++c) {
      sax += r_ax[c][lane];
      say += r_ay[c][lane];
      spe += r_pe[c][lane];
    }
    const float G = *Gp;
    const float mi = bi.z;
    const float2 vi = v_prev[i];
    float vx = vi.x, vy = vi.y;
    if (do_update) {                       // semi-implicit Euler
      vx = fmaf(DT * G, sax, vx);
      vy = fmaf(DT * G, say, vy);
      const float nx = fmaf(DT, vx, xi);
      const float ny = fmaf(DT, vy, yi);
      float4 np; np.x = nx; np.y = ny; np.z = mi; np.w = 0.f;
      pxm_next[i] = np;
      float2 nv; nv.x = vx; nv.y = vy;
      v_next[i] = nv;
      float2* row2 = (float2*)row_out;
      float2 p; p.x = nx; p.y = ny;
      row2[2 * i] = p;
      row2[2 * i + 1] = nv;
    }
    float ke  = 0.5f * mi * (vx * vx + vy * vy);
    float pei = mi * spe;                  // m_i * sum_j m_j/d_ij (over x_{s-1})
    #pragma unroll
    for (int off = 16; off > 0; off >>= 1) {
      ke  += __shfl_down(ke, off, 32);
      pei += __shfl_down(pei, off, 32);
    }
    if (lane == 0) {
      if (do_update) partKE_s[blockIdx.x] = ke;
      partPE_sm1[blockIdx.x] = pei;
    }
  }
}

// ---- finalize: ke_s / pe_s = [256x64 partials] x ones(64), via WMMA ----
typedef float v2f __attribute__((ext_vector_type(2)));
typedef float v8f __attribute__((ext_vector_type(8)));

__global__ __launch_bounds__(512) void grav_finalize(
    const float* __restrict__ partKE, const float* __restrict__ partPE,
    const float* __restrict__ Gp, float* __restrict__ out)
{
  const int lane = threadIdx.x & 31;
  const int wave = threadIdx.x >> 5;       // 16 waves, one 16-step tile each
  const int mrow = lane & 15;              // step row within tile (A: M=lane%16)
  const int half = lane >> 4;              // A K-halves: lanes 0-15 K=0,1; 16-31 K=2,3
  const int step0 = wave * 16;
#if __has_builtin(__builtin_amdgcn_wmma_f32_16x16x4_f32)
  const v2f ones = {1.0f, 1.0f};
  v8f acc_ke = {};
  v8f acc_pe = {};
  #pragma unroll
  for (int c = 0; c < 16; ++c) {           // K = 64 partials in 16 chunks of 4
    const int col = 4 * c + 2 * half;
    const float* pk = partKE + (size_t)(step0 + mrow) * NBLK + col;
    const float* pp = partPE + (size_t)(step0 + mrow) * NBLK + col;
    v2f a_ke; a_ke[0] = pk[0]; a_ke[1] = pk[1];
    v2f a_pe; a_pe[0] = pp[0]; a_pe[1] = pp[1];
    acc_ke = __builtin_amdgcn_wmma_f32_16x16x4_f32(
        false, a_ke, false, ones, (short)0, acc_ke, false, false);
    acc_pe = __builtin_amdgcn_wmma_f32_16x16x4_f32(
        false, a_pe, false, ones, (short)0, acc_pe, false, false);
  }
  const float G = *Gp;
  // D layout: VGPR r = row M=r (lanes 0-15) / M=8+r (lanes 16-31); all N equal.
  if (mrow == 0) {                         // lanes 0 and 16 cover rows 0..15
    #pragma unroll
    for (int r = 0; r < 8; ++r) {
      const int s = step0 + half * 8 + r;
      const float ke = acc_ke[r];
      const float gpe = -0.5f * G * acc_pe[r];
      float* row = out + (size_t)s * ROWF;
      row[NB * 4]     = ke + gpe;          // Hamiltonian
      row[NB * 4 + 1] = gpe;
    }
  }
#else
  // Scalar fallback (one lane per step in this tile).
  if (half == 0) {
    const int s = step0 + mrow;
    float ke = 0.f, pe = 0.f;
    for (int b = 0; b < NBLK; ++b) {
      ke += partKE[(size_t)s * NBLK + b];
      pe += partPE[(size_t)s * NBLK + b];
    }
    const float G = *Gp;
    const float gpe = -0.5f * G * pe;
    float* row = out + (size_t)s * ROWF;
    row[NB * 4]     = ke + gpe;
    row[NB * 4 + 1] = gpe;
  }
#endif
}

extern "C" void kernel_launch(void* const* d_in, const int* in_sizes, int n_in,
                              void* d_out, int out_size, void* d_ws, size_t ws_size,
                              hipStream_t stream) {
  (void)in_sizes; (void)n_in; (void)out_size; (void)ws_size;
  const float* m  = (const float*)d_in[0];
  const float* x0 = (const float*)d_in[1];
  const float* v0 = (const float*)d_in[2];
  const float* Gp = (const float*)d_in[3];   // read on device (graph-safe)
  // d_in[4] = steps (256 per setup_inputs); fixed at compile time.
  float* out = (float*)d_out;

  // Workspace (16B-aligned slabs).
  float4* pxm0   = (float4*)d_ws;            // 32 KB packed (x,y,m)
  float4* pxm1   = pxm0 + NB;                // 32 KB
  float2* vb0    = (float2*)(pxm1 + NB);     // 16 KB velocities
  float2* vb1    = vb0 + NB;                 // 16 KB
  float*  partKE = (float*)(vb1 + NB);       // STEPS*NBLK floats = 64 KB
  float*  partPE = partKE + STEPS * NBLK;    // 64 KB

  grav_init<<<NBLK, 32, 0, stream>>>(x0, v0, m, out, pxm0, vb0, partKE);

  for (int s = 1; s < STEPS; ++s) {
    const bool odd = (s & 1) != 0;
    grav_step<<<NBLK, TPB, 0, stream>>>(
        odd ? pxm0 : pxm1, odd ? pxm1 : pxm0,
        odd ? vb0 : vb1,   odd ? vb1 : vb0,
        out + (size_t)s * ROWF, Gp,
        partKE + s * NBLK, partPE + (size_t)(s - 1) * NBLK, /*do_update=*/1);
  }
  // Final GPE-only pass over x_{STEPS-1} (no state update, no KE write).
  const bool lodd = ((STEPS - 1) & 1) != 0;
  grav_step<<<NBLK, TPB, 0, stream>>>(
      lodd ? pxm1 : pxm0, pxm0,
      lodd ? vb1 : vb0, vb0,
      out, Gp,
      partKE, partPE + (size_t)(STEPS - 1) * NBLK, /*do_update=*/0);

  grav_finalize<<<1, 512, 0, stream>>>(partKE, partPE, Gp, out);
}